// VectorQuantizerEMA_58188216926435
// MI455X (gfx1250) — compile-verified
//
#include <hip/hip_runtime.h>
#include <hip/hip_bf16.h>

#define K_CODES 1024
#define D_DIM   64
#define N_TOTAL (16 * 64 * 64)   // 65536 vectors
#define HW      4096             // 64*64 spatial
#define DECAY_C 0.99f
#define OMD_C   0.01f
#define EPS_C   1e-5f

typedef __attribute__((ext_vector_type(2))) float v2f;
typedef __attribute__((ext_vector_type(8))) float v8f;

// ---- CDNA5 async global->LDS staging (ASYNCcnt path), with safe fallback ----
#if defined(__AMDGCN__) && __has_builtin(__builtin_amdgcn_global_load_async_to_lds_b32)
#define VQ_ASYNC 1
#else
#define VQ_ASYNC 0
#endif

#define AS_GLOBAL __attribute__((address_space(1)))
#define AS_LDS    __attribute__((address_space(3)))

__device__ __forceinline__ void vq_async_copy_b32(const float* src, float* dst_lds) {
#if VQ_ASYNC
    __builtin_amdgcn_global_load_async_to_lds_b32(
        (AS_GLOBAL int*)src, (AS_LDS int*)dst_lds, 0, 0);
#else
    *dst_lds = *src;
#endif
}

__device__ __forceinline__ void vq_async_wait0() {
#if VQ_ASYNC
#if __has_builtin(__builtin_amdgcn_s_wait_asynccnt)
    __builtin_amdgcn_s_wait_asynccnt(0);
#else
    asm volatile("s_wait_asynccnt 0x0" ::: "memory");
#endif
#endif
}

// z_e is [B=16, D=64, H=64, W=64]; flat vector id n = b*4096 + h*64 + w.
// element (n, d) lives at (n>>12)*262144 + d*4096 + (n&4095).

// ---------------------------------------------------------------------------
// Kernel 0: per-code squared norms; zero histogram + segment-sum scratch.
// ---------------------------------------------------------------------------
__global__ void vq_prep(const float* __restrict__ emb,
                        float* __restrict__ embnorm,
                        float* __restrict__ counts,
                        float* __restrict__ embupd) {
    const int tid = blockIdx.x * blockDim.x + threadIdx.x;
    const int nthreads = gridDim.x * blockDim.x;
    if (tid < K_CODES) {
        float s = 0.0f;
        #pragma unroll 8
        for (int d = 0; d < D_DIM; ++d) {
            float e = emb[(size_t)d * K_CODES + tid];
            s += e * e;
        }
        embnorm[tid] = s;
        counts[tid] = 0.0f;
    }
    for (int i = tid; i < D_DIM * K_CODES; i += nthreads) embupd[i] = 0.0f;
}

// ---------------------------------------------------------------------------
// Kernel 1: WMMA fp32 distance matmul + argmin + gather + EMA scatter.
// 256 threads = 8 waves; each wave owns a 16-row z tile; code tiles are
// double-buffered in LDS via async global->LDS copies and shared by all
// 8 waves (8x less L2 traffic, B fragments via conflict-free ds_load_b64).
// ---------------------------------------------------------------------------
__global__ __launch_bounds__(256)
void vq_dist_argmin(const float* __restrict__ z,
                    const float* __restrict__ emb,
                    const float* __restrict__ embnorm,
                    float* __restrict__ zq_out,
                    float* __restrict__ idx_out,
                    float* __restrict__ counts,
                    float* __restrict__ embupd) {
    __shared__ float sB[2][16][68];     // [buf][code][d], pad 68 -> 0 bank conflicts
    __shared__ float sNorm[K_CODES];
    __shared__ float sMin[8][16][16];
    __shared__ int   sIdx[8][16][16];
    __shared__ int   sBest[8][16];

    const int tid   = threadIdx.x;
    const int wave  = tid >> 5;
    const int lane  = tid & 31;
    const int col   = lane & 15;        // A-row / C-column owned by this lane
    const int khalf = lane >> 4;        // ISA A 16x4 layout: lanes16-31 hold K=2,3
    const int n0    = (blockIdx.x * 8 + wave) * 16;

    // Stage ||e_k||^2 into LDS once per block.
    for (int i = tid; i < K_CODES; i += 256) sNorm[i] = embnorm[i];

    // Preload A fragments: the wave's whole 16x64 z tile lives in VGPRs.
    v2f a[16];
    const size_t zbase = (size_t)(n0 >> 12) * 262144u + (size_t)(n0 & 4095);
    {
        const size_t rbase = zbase + (size_t)col;   // row n0+col
        #pragma unroll
        for (int s = 0; s < 16; ++s) {
            const int d0 = 4 * s + 2 * khalf;
            a[s].x = z[rbase + (size_t)d0 * HW];
            a[s].y = z[rbase + (size_t)(d0 + 1) * HW];
        }
    }

    // Stage first code tile (transposed: sB[c][d] = emb[d*K + k0 + c]).
    #pragma unroll
    for (int i = 0; i < 4; ++i) {
        const int e = tid + 256 * i;            // 0..1023
        const int d = e >> 4, c = e & 15;       // coalesced 64B rows of emb
        vq_async_copy_b32(emb + (size_t)d * K_CODES + c, &sB[0][c][d]);
    }

    float minv[8];
    int   mini[8];
    #pragma unroll
    for (int j = 0; j < 8; ++j) { minv[j] = 3.4e38f; mini[j] = 0; }

    for (int t = 0; t < K_CODES / 16; ++t) {
        vq_async_wait0();          // own tile-t copies done
        __syncthreads();           // everyone's copies done; prev compute done

        if (t + 1 < K_CODES / 16) {
            const int k0n = (t + 1) * 16;
            float (*dbuf)[68] = sB[(t + 1) & 1];
            #pragma unroll
            for (int i = 0; i < 4; ++i) {
                const int e = tid + 256 * i;
                const int d = e >> 4, c = e & 15;
                vq_async_copy_b32(emb + (size_t)d * K_CODES + k0n + c, &dbuf[c][d]);
            }
        }

        const int k0 = t * 16;
        const float* bp = &sB[t & 1][col][0];
        v8f acc = {};
        #pragma unroll
        for (int s = 0; s < 16; ++s) {
            const int d0 = 4 * s + 2 * khalf;   // B 4x16 rows mirror A layout
            v2f bf = *(const v2f*)(bp + d0);    // one ds_load_b64, conflict-free
            acc = __builtin_amdgcn_wmma_f32_16x16x4_f32(
                false, a[s], false, bf, (short)0, acc, false, false);
        }
        const float en = sNorm[k0 + col];
        #pragma unroll
        for (int j = 0; j < 8; ++j) {
            const float dist = en - 2.0f * acc[j];  // ||z||^2 omitted: row-const
            if (dist < minv[j]) { minv[j] = dist; mini[j] = k0 + col; }
        }
    }

    // Cross-lane argmin per row via LDS (ties -> lowest index, like jnp.argmin).
    #pragma unroll
    for (int j = 0; j < 8; ++j) {
        const int m = j + 8 * khalf;   // C layout: VGPR j -> M = j + 8*(lane>=16)
        sMin[wave][m][col] = minv[j];
        sIdx[wave][m][col] = mini[j];
    }
    __syncthreads();

    if (lane < 16) {
        float best = sMin[wave][lane][0];
        int   bidx = sIdx[wave][lane][0];
        #pragma unroll
        for (int c = 1; c < 16; ++c) {
            const float v = sMin[wave][lane][c];
            const int   i = sIdx[wave][lane][c];
            if (v < best || (v == best && i < bidx)) { best = v; bidx = i; }
        }
        sBest[wave][lane] = bidx;
        idx_out[n0 + lane] = (float)bidx;
        atomicAdd(&counts[bidx], 1.0f);           // one-hot histogram
    }
    __syncthreads();

    // EMA segment-sum scatter straight from the A fragments (no z re-read).
    {
        const int bi = sBest[wave][col];          // this lane's row == col
        #pragma unroll
        for (int s = 0; s < 16; ++s) {
            const int d0 = 4 * s + 2 * khalf;
            atomicAdd(&embupd[(size_t)d0 * K_CODES + bi], a[s].x);
            atomicAdd(&embupd[(size_t)(d0 + 1) * K_CODES + bi], a[s].y);
        }
    }

    // Gather z_q = emb[:, idx]; for fixed d the 16 tile rows are contiguous
    // in [B,D,H,W], so consecutive lanes produce coalesced 64B stores.
    for (int i = lane; i < 16 * D_DIM; i += 32) {
        const int d = i >> 4;
        const int r = i & 15;
        zq_out[zbase + (size_t)d * HW + r] =
            emb[(size_t)d * K_CODES + sBest[wave][r]];
    }
}

// ---------------------------------------------------------------------------
// Kernel 2: EMA count/codebook update + normalization. One 1024-thread block.
// ---------------------------------------------------------------------------
__global__ __launch_bounds__(1024)
void vq_finalize(const float* __restrict__ emb_ema,
                 const float* __restrict__ counts_ema,
                 const float* __restrict__ counts,
                 const float* __restrict__ embupd,
                 float* __restrict__ out_emb,
                 float* __restrict__ out_emb_ema,
                 float* __restrict__ out_counts) {
    __shared__ float red[K_CODES];
    const int k = threadIdx.x;

    const float c_new = DECAY_C * counts_ema[k] + OMD_C * counts[k];
    out_counts[k] = c_new;
    red[k] = c_new;
    __syncthreads();
    for (int s = 512; s > 0; s >>= 1) {
        if (k < s) red[k] += red[k + s];
        __syncthreads();
    }
    const float n = red[0];
    const float denom = (c_new + EPS_C) / (n + (float)K_CODES * EPS_C) * n;
    const float inv = 1.0f / denom;

    for (int d = 0; d < D_DIM; ++d) {
        const int i = d * K_CODES + k;            // coalesced over k
        const float ema = DECAY_C * emb_ema[i] + OMD_C * embupd[i];
        out_emb_ema[i] = ema;
        out_emb[i] = ema * inv;
    }
}

// ---------------------------------------------------------------------------
extern "C" void kernel_launch(void* const* d_in, const int* in_sizes, int n_in,
                              void* d_out, int out_size, void* d_ws, size_t ws_size,
                              hipStream_t stream) {
    const float* z_e        = (const float*)d_in[0];  // [16,64,64,64]
    const float* emb        = (const float*)d_in[1];  // [64,1024]
    const float* emb_ema    = (const float*)d_in[2];  // [64,1024]
    const float* counts_ema = (const float*)d_in[3];  // [1024]

    // Outputs, concatenated flat in reference return order (all float):
    float* out         = (float*)d_out;
    float* zq_out      = out;                               // 4194304
    float* idx_out     = zq_out + (size_t)N_TOTAL * D_DIM;  // 65536
    float* emb_out     = idx_out + N_TOTAL;                 // 65536
    float* emb_ema_out = emb_out + D_DIM * K_CODES;         // 65536
    float* counts_out  = emb_ema_out + D_DIM * K_CODES;     // 1024

    // Workspace: embnorm[1024] | counts[1024] | embupd[64*1024]  (264 KB)
    float* ws      = (float*)d_ws;
    float* embnorm = ws;
    float* counts  = ws + K_CODES;
    float* embupd  = ws + 2 * K_CODES;

    vq_prep<<<64, 256, 0, stream>>>(emb, embnorm, counts, embupd);
    vq_dist_argmin<<<N_TOTAL / (16 * 8), 256, 0, stream>>>(
        z_e, emb, embnorm, zq_out, idx_out, counts, embupd);
    vq_finalize<<<1, 1024, 0, stream>>>(
        emb_ema, counts_ema, counts, embupd, emb_out, emb_ema_out, counts_out);
}